// VectorQuantizerEMA_66005057405363
// MI455X (gfx1250) — compile-verified
//
#include <hip/hip_runtime.h>
#include <hip/hip_bf16.h>

typedef __attribute__((ext_vector_type(2))) float v2f;
typedef __attribute__((ext_vector_type(8))) float v8f;

#define NUM_EMBED 512
#define EMBED_DIM 64
#define NB 32
#define NH 64
#define NW 64
#define NTOK (NB * NH * NW)          // 131072

// d_out flat offsets (reference tuple order, all fp32)
#define OFF_LOSS     0
#define OFF_OUT      1
#define OFF_PERP     8388609
#define OFF_ENC      8388610
#define OFF_NEWEMB   75497474
#define OFF_CLUSTER  75530242
#define OFF_NEWEMAW  75530754

// workspace layout (floats)
#define WS_COUNTS 0          // 512
#define WS_DW     512        // 512*64
#define WS_SSE    (512 + 512*64)
#define WS_FLOATS (512 + 512*64 + 1)

// LDS layout (floats), embedding padded to stride 68 (bank = (4e+d)%64)
#define EMB_STRIDE 68
#define S_EMB   0
#define S_E2    (NUM_EMBED * EMB_STRIDE)             // 34816
#define S_IDX   (S_E2 + NUM_EMBED)                   // 35328 (64 ints)
#define S_RED   (S_IDX + 64)                         // 35392 (128 floats)
#define SMEM_FLOATS (S_RED + 128)                    // 35520
#define SMEM_BYTES (SMEM_FLOATS * 4)                 // 142080

__global__ __launch_bounds__(128) void vq_main_kernel(
    const float* __restrict__ x, const float* __restrict__ emb,
    float* __restrict__ out, float* __restrict__ counts,
    float* __restrict__ dw, float* __restrict__ sse) {
  extern __shared__ float smem[];
  float* emb_s = smem + S_EMB;
  float* e2_s  = smem + S_E2;
  int*   idx_s = (int*)(smem + S_IDX);
  float* red_s = smem + S_RED;

  const int tid = threadIdx.x;
  const int bh  = blockIdx.x;          // 0..2047
  const int b   = bh >> 6;
  const int h   = bh & 63;

  // ---- stage codebook into LDS (padded rows) ----
  for (int i = tid; i < NUM_EMBED * EMBED_DIM; i += 128) {
    int e = i >> 6, d = i & 63;
    emb_s[e * EMB_STRIDE + d] = emb[i];
  }
  __syncthreads();
  // per-code squared norms
  for (int e = tid; e < NUM_EMBED; e += 128) {
    const float* r = emb_s + e * EMB_STRIDE;
    float s = 0.f;
    #pragma unroll
    for (int d = 0; d < EMBED_DIM; ++d) s += r[d] * r[d];
    e2_s[e] = s;
  }
  __syncthreads();

  const int lane = tid & 31;
  const int wave = tid >> 5;           // 0..3 : token tile (16 tokens each)
  const int l16  = lane & 15;
  const int hi   = lane >> 4;          // half-wave selects K pair / M+8

  // ---- load A fragments: 16 tokens x 64 features (fp32, K=4 per WMMA) ----
  const int wtok = wave * 16 + l16;    // token's w coordinate
  const float* xb = x + (size_t)b * (64 * 64 * 64) + h * 64 + wtok;
  v2f a[16];
  #pragma unroll
  for (int k = 0; k < 16; ++k) {
    int c0 = k * 4 + 2 * hi;           // channel (= feature dim), stride 4096 floats
    a[k].x = xb[(size_t)c0 * 4096];
    a[k].y = xb[(size_t)(c0 + 1) * 4096];
  }

  float best[8];
  int   bidx[8];
  #pragma unroll
  for (int v = 0; v < 8; ++v) { best[v] = 3.4e38f; bidx[v] = 0; }

  // ---- sweep 32 code tiles of 16; D = A(16x64) x B(64x16) via 16 chained WMMAs ----
  for (int j = 0; j < 32; ++j) {
    const int code = j * 16 + l16;     // this lane's B column / e2 code
    const float* brow = emb_s + code * EMB_STRIDE + 2 * hi;  // 8B aligned (stride 68)
    v8f acc = {};
    #pragma unroll
    for (int k = 0; k < 16; ++k) {
      v2f bf = *(const v2f*)(brow + k * 4);
      acc = __builtin_amdgcn_wmma_f32_16x16x4_f32(
          false, a[k], false, bf, (short)0, acc, false, false);
    }
    const float e2v = e2_s[code];
    #pragma unroll
    for (int v = 0; v < 8; ++v) {
      float dval = e2v - 2.0f * acc[v];          // ||e||^2 - 2 x.e (argmin-equivalent)
      if (dval < best[v] || (dval == best[v] && code < bidx[v])) {
        best[v] = dval; bidx[v] = code;
      }
    }
  }

  // ---- argmin across the 16 lanes holding each row (xor stays within half) ----
  #pragma unroll
  for (int m = 1; m < 16; m <<= 1) {
    #pragma unroll
    for (int v = 0; v < 8; ++v) {
      float ov = __shfl_xor(best[v], m, 32);
      int   oi = __shfl_xor(bidx[v], m, 32);
      if (ov < best[v] || (ov == best[v] && oi < bidx[v])) {
        best[v] = ov; bidx[v] = oi;
      }
    }
  }
  if (l16 == 0) {
    #pragma unroll
    for (int v = 0; v < 8; ++v) idx_s[wave * 16 + hi * 8 + v] = bidx[v];
  }
  __syncthreads();

  // ---- cluster counts (exact: fp32 adds of 1.0) ----
  if (tid < 64) atomicAdd(&counts[idx_s[tid]], 1.0f);

  // ---- quantized output (NCHW), SSE loss partial, dw scatter ----
  float lsum = 0.f;
  for (int i = 0; i < 32; ++i) {
    int p = tid + i * 128;             // 4096 (d,w) pairs
    int w = p & 63, d = p >> 6;
    int e = idx_s[w];
    float q  = emb_s[e * EMB_STRIDE + d];
    size_t oix = (((size_t)(b * 64 + d) * 64) + h) * 64 + w;  // NCHW index
    float xv = x[oix];
    __builtin_nontemporal_store(q, &out[OFF_OUT + oix]);
    float df = q - xv;
    lsum += df * df;
    atomicAdd(&dw[e * 64 + d], xv);
  }
  red_s[tid] = lsum;
  __syncthreads();
  for (int s = 64; s > 0; s >>= 1) {
    if (tid < s) red_s[tid] += red_s[tid + s];
    __syncthreads();
  }
  if (tid == 0) atomicAdd(sse, red_s[0]);

  // ---- one-hot encodings: 64 contiguous rows of 512 (single-pass predicate fill) ----
  const int n0 = bh * 64;              // first token of this WG
  v2f* enc = (v2f*)(out + OFF_ENC) + (size_t)n0 * 256;  // 256 v2f per row
  for (int i = 0; i < 128; ++i) {
    int c = tid + i * 128;             // 16384 v2f chunks
    int row = c >> 8, col2 = c & 255;
    int e = idx_s[row];
    v2f v = {};
    if ((e >> 1) == col2) { if (e & 1) v.y = 1.0f; else v.x = 1.0f; }
    __builtin_nontemporal_store(v, &enc[(size_t)row * 256 + col2]);
  }
}

__global__ __launch_bounds__(512) void vq_finalize_kernel(
    const float* __restrict__ ema_w, const float* __restrict__ ema_cs,
    const float* __restrict__ counts, const float* __restrict__ dw,
    const float* __restrict__ sse, float* __restrict__ out) {
  __shared__ float cl_s[512];
  __shared__ float red[512];
  const int t = threadIdx.x;

  float cnt  = counts[t];
  float craw = ema_cs[t] * 0.99f + 0.01f * cnt;

  // k = sum(craw)
  red[t] = craw;
  __syncthreads();
  for (int s = 256; s > 0; s >>= 1) { if (t < s) red[t] += red[t + s]; __syncthreads(); }
  float k = red[0];
  __syncthreads();

  float cl = (craw + 1e-5f) / (k + 512.0f * 1e-5f) * k;  // Laplace smoothing
  cl_s[t] = cl;
  out[OFF_CLUSTER + t] = cl;

  // perplexity = exp(-sum(p log(p+1e-10))), p = counts/N
  float p = cnt * (1.0f / (float)NTOK);
  red[t] = p * logf(p + 1e-10f);
  __syncthreads();
  for (int s = 256; s > 0; s >>= 1) { if (t < s) red[t] += red[t + s]; __syncthreads(); }
  if (t == 0) {
    out[OFF_PERP] = expf(-red[0]);
    out[OFF_LOSS] = 0.25f * sse[0] / ((float)NTOK * (float)EMBED_DIM);
  }
  __syncthreads();

  // new_ema_w = 0.99*ema_w + 0.01*dw ; new_embedding = new_ema_w / cluster
  for (int i = t; i < NUM_EMBED * EMBED_DIM; i += 512) {
    int e = i >> 6;
    float nw = ema_w[i] * 0.99f + 0.01f * dw[i];
    out[OFF_NEWEMAW + i] = nw;
    out[OFF_NEWEMB  + i] = nw / cl_s[e];
  }
}

extern "C" void kernel_launch(void* const* d_in, const int* in_sizes, int n_in,
                              void* d_out, int out_size, void* d_ws, size_t ws_size,
                              hipStream_t stream) {
  const float* x      = (const float*)d_in[0];
  const float* emb    = (const float*)d_in[1];
  const float* ema_w  = (const float*)d_in[2];
  const float* ema_cs = (const float*)d_in[3];
  float* out = (float*)d_out;

  float* ws     = (float*)d_ws;
  float* counts = ws + WS_COUNTS;
  float* dw     = ws + WS_DW;
  float* sse    = ws + WS_SSE;

  (void)hipMemsetAsync(d_ws, 0, WS_FLOATS * sizeof(float), stream);

  dim3 grid(NB * NH);   // 2048 workgroups: one per (b,h) -> 64 tokens each
  dim3 block(128);      // 4 waves x 16-token tiles
  vq_main_kernel<<<grid, block, SMEM_BYTES, stream>>>(x, emb, out, counts, dw, sse);
  vq_finalize_kernel<<<1, 512, 0, stream>>>(ema_w, ema_cs, counts, dw, sse, out);
}